// Attention_8529805050365
// MI455X (gfx1250) — compile-verified
//
#include <hip/hip_runtime.h>
#include <math.h>

// Problem constants (from reference): B=32, S=2048, H=1024
#define B_ 32
#define S_ 2048
#define H_ 1024
#define PITCH 68   // LDS row pitch in floats: 64 data + 4 pad -> 272B rows (16B aligned), no 64-bank conflicts

typedef float v2f __attribute__((ext_vector_type(2)));
typedef float v8f __attribute__((ext_vector_type(8)));
typedef int   i32x4 __attribute__((ext_vector_type(4)));

// ---------------------------------------------------------------------------
// CDNA5 async global->LDS copy path (ASYNCcnt-tracked). Guarded so a builtin
// name mismatch falls back to synchronous staging instead of failing compile.
// Probe-learned signature: param0 = AS1 (global) i32x4*, param1 = AS3 (LDS).
// ---------------------------------------------------------------------------
#if __has_builtin(__builtin_amdgcn_global_load_async_to_lds_b128)
#define HAVE_ASYNC_LDS 1
typedef __attribute__((address_space(1))) i32x4 g_i32x4;
typedef __attribute__((address_space(3))) i32x4 l_i32x4;
#else
#define HAVE_ASYNC_LDS 0
#endif

__device__ __forceinline__ void copy_f4_to_lds(const float* g, float* l) {
#if HAVE_ASYNC_LDS
    __builtin_amdgcn_global_load_async_to_lds_b128(
        (g_i32x4*)g, (l_i32x4*)l, 0, 0);
#else
    float4 f4 = *(const float4*)g;
    l[0] = f4.x; l[1] = f4.y; l[2] = f4.z; l[3] = f4.w;
#endif
}

template <int N>
__device__ __forceinline__ void async_wait() {
#if HAVE_ASYNC_LDS
#if __has_builtin(__builtin_amdgcn_s_wait_asynccnt)
    __builtin_amdgcn_s_wait_asynccnt(N);
#else
    asm volatile("s_wait_asynccnt %0" :: "i"(N) : "memory");
#endif
#endif
}

// ---------------------------------------------------------------------------
// Kernel 1: q[b,k] = sum_h hidden[b,h] * W1[k,h] + attn_b[k]
//   W1[k,h] = attn_W[k*2H + h]. 67 MFLOP total -> negligible.
// ---------------------------------------------------------------------------
__global__ __launch_bounds__(256) void qproj_kernel(const float* __restrict__ hidden,
                                                    const float* __restrict__ attn_W,
                                                    const float* __restrict__ attn_b,
                                                    float* __restrict__ q) {
    int t = blockIdx.x * blockDim.x + threadIdx.x;   // 0 .. B*H-1
    int b = t >> 10;
    int k = t & (H_ - 1);
    const float4* hp = (const float4*)(hidden + (size_t)b * H_);
    const float4* wp = (const float4*)(attn_W + (size_t)k * (2 * H_));
    float s = 0.f;
    #pragma unroll 4
    for (int i = 0; i < H_ / 4; ++i) {
        float4 h4 = hp[i];
        float4 w4 = wp[i];
        s += h4.x * w4.x + h4.y * w4.y + h4.z * w4.z + h4.w * w4.w;
    }
    q[t] = s + attn_b[k];
}

// ---------------------------------------------------------------------------
// Issue a ROWSx64 f32 tile copy (global, row stride `stride`) into LDS pitch
// PITCH. 128 threads: 8 groups of 16 lanes; each group owns ROWS/8 rows,
// lanes cover the 16 float4 columns of a row (256B coalesced per group).
// ROWS=128 -> 16 b128 issues/thread; ROWS=64 -> 8.
// ---------------------------------------------------------------------------
template <int ROWS>
__device__ __forceinline__ void stage_issue(float* dst, const float* srcBase,
                                            size_t stride, int tid) {
    const int g = tid >> 4;       // 0..7
    const int c = tid & 15;       // float4 column
    constexpr int RPG = ROWS / 8; // rows per group
    #pragma unroll
    for (int rr = 0; rr < RPG; ++rr) {
        int row = g * RPG + rr;
        copy_f4_to_lds(srcBase + (size_t)row * stride + c * 4, dst + row * PITCH + c * 4);
    }
}

// ---------------------------------------------------------------------------
// Kernel 2: main compute. Workgroup = 128 threads (4 waves), tile = (b, 128 s).
//   for kt (16 tiles of 64 k):
//     acc: per wave 32s x 64k = 8 v8f frags (two 16-row M-subtiles x 4 k-frags)
//     for ht (16 panels of 64 h), double-buffered async LDS staging:
//       prefetch panel ht+1; s_wait_asynccnt 24; barrier; 128 WMMAs on panel ht
//     score[s] += sum_k v[k] * tanh(acc + q[b,k])
// W2[k,h] = attn_W[k*2H + H + h].
// Per wave: 16kt x 16ht x 16 K-steps x 8 = 32768 v_wmma_f32_16x16x4_f32.
// ---------------------------------------------------------------------------
__global__ __launch_bounds__(128) void energy_kernel(const float* __restrict__ enc,
                                                     const float* __restrict__ attn_W,
                                                     const float* __restrict__ qb,
                                                     const float* __restrict__ v,
                                                     float* __restrict__ scores) {
    __shared__ float encT[2][128 * PITCH];   // 2 x 34.8 KB
    __shared__ float w2T[2][64 * PITCH];     // 2 x 17.4 KB
    __shared__ float sacc[128];

    const int b   = blockIdx.y;
    const int s0  = blockIdx.x * 128;
    const int tid = threadIdx.x;
    const int w    = tid >> 5;        // wave 0..3 -> owns s-rows [w*32, w*32+32)
    const int lane = tid & 31;
    const int kl   = lane & 15;       // N / column index within 16
    const int half = lane >> 4;       // 0 or 1

    sacc[tid] = 0.f;

    const float* encBlock = enc + ((size_t)b * S_ + s0) * H_;

    for (int kt = 0; kt < H_ / 64; ++kt) {
        v8f accA[4] = {{}, {}, {}, {}};   // M-subtile 0 (rows w*32 + 0..15)
        v8f accB[4] = {{}, {}, {}, {}};   // M-subtile 1 (rows w*32 + 16..31)
        const int kbase = kt * 64;
        const float* w2Block = attn_W + (size_t)kbase * (2 * H_) + H_;

        // Preload panel ht=0 into buffer 0 (16 + 8 = 24 async issues per thread)
        stage_issue<128>(encT[0], encBlock, (size_t)H_, tid);
        stage_issue<64>(w2T[0], w2Block, (size_t)(2 * H_), tid);

        int cur = 0;
        for (int ht = 0; ht < H_ / 64; ++ht) {
            // Prefetch next panel into the other buffer while this one computes
            if (ht + 1 < H_ / 64) {
                const int hb = (ht + 1) * 64;
                stage_issue<128>(encT[cur ^ 1], encBlock + hb, (size_t)H_, tid);
                stage_issue<64>(w2T[cur ^ 1], w2Block + hb, (size_t)(2 * H_), tid);
                async_wait<24>();   // leave only the prefetch outstanding
            } else {
                async_wait<0>();
            }
            __syncthreads();        // all waves' panel-ht data resident

            const float* eb = encT[cur];
            const float* wb = w2T[cur];
            const float* aRow0 = eb + (w * 32 + kl) * PITCH;
            const float* aRow1 = aRow0 + 16 * PITCH;
            #pragma unroll
            for (int hh = 0; hh < 64; hh += 4) {
                // A 16x4 f32 layout: lanes<16 hold K={hh,hh+1}, lanes>=16 hold K={hh+2,hh+3}
                v2f a0, a1;
                a0.x = aRow0[hh + half * 2];
                a0.y = aRow0[hh + half * 2 + 1];
                a1.x = aRow1[hh + half * 2];
                a1.y = aRow1[hh + half * 2 + 1];
                // B 4x16 layout: VGPR0 rows h={0,1}, VGPR1 rows h={2,3}; cols on lane&15
                #pragma unroll
                for (int kf = 0; kf < 4; ++kf) {
                    const float* bRow = wb + (kf * 16 + kl) * PITCH;
                    v2f bm;
                    bm.x = bRow[hh + half];
                    bm.y = bRow[hh + half + 2];
                    accA[kf] = __builtin_amdgcn_wmma_f32_16x16x4_f32(
                        false, a0, false, bm, (short)0, accA[kf], false, false);
                    accB[kf] = __builtin_amdgcn_wmma_f32_16x16x4_f32(
                        false, a1, false, bm, (short)0, accB[kf], false, false);
                }
            }
            __syncthreads();   // panel fully consumed -> next iteration may overwrite it
            cur ^= 1;
        }

        // Epilogue for this k-tile: e = tanh(acc + q), score += e * v[k]
        float qv[4], vv[4];
        #pragma unroll
        for (int kf = 0; kf < 4; ++kf) {
            qv[kf] = qb[(size_t)b * H_ + kbase + kf * 16 + kl];
            vv[kf] = v[kbase + kf * 16 + kl];
        }

        #pragma unroll
        for (int r = 0; r < 8; ++r) {
            // C layout: VGPR r, lanes0-15 -> M=r, lanes16-31 -> M=r+8; N = lane&15
            float pA = 0.f, pB = 0.f;
            #pragma unroll
            for (int kf = 0; kf < 4; ++kf) {
                pA += tanhf(accA[kf][r] + qv[kf]) * vv[kf];
                pB += tanhf(accB[kf][r] + qv[kf]) * vv[kf];
            }
            // reduce over the 16 N-lanes within each half
            pA += __shfl_xor(pA, 8, 32);  pB += __shfl_xor(pB, 8, 32);
            pA += __shfl_xor(pA, 4, 32);  pB += __shfl_xor(pB, 4, 32);
            pA += __shfl_xor(pA, 2, 32);  pB += __shfl_xor(pB, 2, 32);
            pA += __shfl_xor(pA, 1, 32);  pB += __shfl_xor(pB, 1, 32);
            if (kl == 0) {
                int srow = w * 32 + r + half * 8;
                sacc[srow]      += pA;   // wave-exclusive s ranges, no races
                sacc[srow + 16] += pB;
            }
        }
    }

    __syncthreads();
    scores[(size_t)b * S_ + s0 + tid] = sacc[tid];
}

// ---------------------------------------------------------------------------
// Kernel 3: softmax over S per batch row.
// ---------------------------------------------------------------------------
__global__ __launch_bounds__(256) void softmax_kernel(const float* __restrict__ scores,
                                                      float* __restrict__ out) {
    __shared__ float red[256];
    const int b = blockIdx.x;
    const float* row = scores + (size_t)b * S_;

    float m = -INFINITY;
    for (int s = threadIdx.x; s < S_; s += 256) m = fmaxf(m, row[s]);
    red[threadIdx.x] = m;
    __syncthreads();
    for (int off = 128; off > 0; off >>= 1) {
        if (threadIdx.x < off) red[threadIdx.x] = fmaxf(red[threadIdx.x], red[threadIdx.x + off]);
        __syncthreads();
    }
    m = red[0];
    __syncthreads();

    float sum = 0.f;
    for (int s = threadIdx.x; s < S_; s += 256) sum += __expf(row[s] - m);
    red[threadIdx.x] = sum;
    __syncthreads();
    for (int off = 128; off > 0; off >>= 1) {
        if (threadIdx.x < off) red[threadIdx.x] += red[threadIdx.x + off];
        __syncthreads();
    }
    float inv = 1.0f / red[0];

    for (int s = threadIdx.x; s < S_; s += 256)
        out[(size_t)b * S_ + s] = __expf(row[s] - m) * inv;
}

// ---------------------------------------------------------------------------
// Launch
// ---------------------------------------------------------------------------
extern "C" void kernel_launch(void* const* d_in, const int* in_sizes, int n_in,
                              void* d_out, int out_size, void* d_ws, size_t ws_size,
                              hipStream_t stream) {
    const float* hidden = (const float*)d_in[0];   // (1,B,H)
    const float* enc    = (const float*)d_in[1];   // (B,S,H)
    const float* attn_W = (const float*)d_in[2];   // (H,2H)
    const float* attn_b = (const float*)d_in[3];   // (H,)
    const float* v      = (const float*)d_in[4];   // (H,)
    float* out = (float*)d_out;                    // (B,1,S)

    float* q      = (float*)d_ws;                  // B*H floats
    float* scores = q + (size_t)B_ * H_;           // B*S floats

    qproj_kernel<<<(B_ * H_) / 256, 256, 0, stream>>>(hidden, attn_W, attn_b, q);

    dim3 g2(S_ / 128, B_);
    energy_kernel<<<g2, 128, 0, stream>>>(enc, attn_W, q, v, scores);

    softmax_kernel<<<B_, 256, 0, stream>>>(scores, out);
}